// ActionVQVAE_38431367365102
// MI455X (gfx1250) — compile-verified
//
#include <hip/hip_runtime.h>

// ---------------------------------------------------------------------------
// ActionVQVAE fused forward for MI455X (gfx1250, wave32, WMMA).
// All five GEMMs on V_WMMA_F32_16X16X4_F32 (exact fp32, matrix pipe).
// Weights + transposed codebook staged in LDS (~295 KB of the 320 KB/WGP).
// Each wave32 owns TWO 16-row tiles (M-blocking): every B-fragment from LDS
// feeds two WMMAs, halving LDS bytes per FLOP in the dominant GEMMs.
// x chunks staged via GLOBAL_LOAD_ASYNC_TO_LDS_B128 (ASYNCcnt-tracked DMA).
// A-fragments live in registers; quantized rows are gathered directly into
// A-fragment registers (no LDS bounce for the gather).
// ---------------------------------------------------------------------------

typedef __attribute__((ext_vector_type(2))) float v2f;
typedef __attribute__((ext_vector_type(8))) float v8f;

#define WMMA_F32(a, b, c) \
  __builtin_amdgcn_wmma_f32_16x16x4_f32(false, (a), false, (b), (short)0, (c), false, false)

__device__ inline v8f zero8() {
  v8f v = {0.f, 0.f, 0.f, 0.f, 0.f, 0.f, 0.f, 0.f};
  return v;
}

constexpr int BATCH = 262144;
constexpr int IN    = 28;    // ACTION_DIM * CHUNK_SIZE
constexpr int HID   = 128;
constexpr int LAT   = 64;
constexpr int NC    = 512;
constexpr int RT    = 2;               // row-tiles per wave (M-blocking)
constexpr int ROWS  = 256;             // rows per block per chunk (8 waves x 32)
constexpr int CHUNKS = BATCH / ROWS;   // 1024
constexpr int GRID   = 512;            // must divide CHUNKS (uniform loop count)

// Fragment conventions (wave32, f32 WMMA 16x16x4):
//   A (16x4):  lane l: m = l&15, half = l>>4; a.x = A[m][kb+2*half], a.y = A[m][kb+2*half+1]
//   B (4x16):  lane l: n = l&15;              b.x = B[kb+2*half][n], b.y = B[kb+2*half+1][n]
//   C/D(16x16):lane l holds col n = l&15; vgpr r holds row (r + 8*half)

__global__ __launch_bounds__(256) void vqvae_fwd(
    const float* __restrict__ xg,
    const float* __restrict__ w1,  const float* __restrict__ b1,
    const float* __restrict__ w2,  const float* __restrict__ b2,
    const float* __restrict__ cb,
    const float* __restrict__ dw1, const float* __restrict__ db1,
    const float* __restrict__ dw2, const float* __restrict__ db2,
    float* __restrict__ recon, float* __restrict__ acc) {
  // ---- LDS (totals ~295 KB; 320 KB/WGP on CDNA5) ----
  __shared__ float sW1 [IN * HID];                                  // enc_w1[k][n]
  __shared__ float sB1 [HID];
  __shared__ float sW2 [HID * LAT];                                 // enc_w2[k][n]
  __shared__ float sB2 [LAT];
  __shared__ float sCBt[LAT * NC];                                  // codebook^T
  __shared__ float sE2 [NC];                                        // ||e||^2
  __shared__ float sDW1[LAT * HID];                                 // dec_w1[k][n]
  __shared__ float sDB1[HID];
  __shared__ float sDW2[HID * 32];                                  // dec_w2 (n pad 28->32)
  __shared__ float sDB2[32];
  __shared__ __attribute__((aligned(16))) float sX  [ROWS * IN];    // current x chunk
  __shared__ float sS  [8 * 16 * 16];                               // per-wave bounce tile
  __shared__ __attribute__((aligned(16))) float sZQ [8 * 16 * LAT]; // per-wave staging
  __shared__ float sZ2 [8 * 32];
  __shared__ int   sIdx[8 * 32];
  __shared__ float sRed[16];

  const int tid  = threadIdx.x;
  const int lane = tid & 31;
  const int wid  = tid >> 5;
  const int m16  = lane & 15;
  const int half = lane >> 4;
  const int koff = half * 2;
  const int rbase = wid * (RT * 16);     // local row base of this wave's tiles

  // ---- stage shared weights (once per block) ----
  for (int i = tid; i < IN * HID; i += 256) sW1[i] = w1[i];
  for (int i = tid; i < HID;      i += 256) sB1[i] = b1[i];
  for (int i = tid; i < HID * LAT; i += 256) sW2[i] = w2[i];
  if (tid < LAT) sB2[tid] = b2[tid];
  for (int i = tid; i < NC * LAT; i += 256) {
    const int c = i >> 6, k = i & 63;
    sCBt[k * NC + c] = cb[i];            // coalesced read, transposed LDS write
  }
  for (int i = tid; i < LAT * HID; i += 256) sDW1[i] = dw1[i];
  for (int i = tid; i < HID;       i += 256) sDB1[i] = db1[i];
  for (int i = tid; i < HID * 32; i += 256) {
    const int k = i >> 5, n = i & 31;
    sDW2[i] = (n < IN) ? dw2[k * IN + n] : 0.f;
  }
  if (tid < 32) sDB2[tid] = (tid < IN) ? db2[tid] : 0.f;
  __syncthreads();
  for (int c = tid; c < NC; c += 256) {  // ||e||^2 per code
    float s = 0.f;
    for (int k = 0; k < LAT; ++k) { const float v = sCBt[k * NC + c]; s += v * v; }
    sE2[c] = s;
  }

  float rlAcc = 0.f, vqAcc = 0.f;

  for (int chunk = blockIdx.x; chunk < CHUNKS; chunk += GRID) {
    __syncthreads();                     // protect sX/sZQ from previous iteration
    const float* xc = xg + (size_t)chunk * (ROWS * IN);
    // ---- async DMA stage of x chunk: global -> LDS, ASYNCcnt-tracked ----
    for (int i = tid; i < (ROWS * IN) / 4; i += 256) {
      const float*   g = xc + i * 4;
      const unsigned l = (unsigned)(size_t)(&sX[i * 4]);  // low 32b = LDS byte addr
      asm volatile("global_load_async_to_lds_b128 %0, %1, off"
                   :: "v"(l), "v"(g) : "memory");
    }
    asm volatile("s_wait_asynccnt 0" ::: "memory");
    if (chunk + GRID < CHUNKS)           // warm L2 for next chunk
      __builtin_prefetch(xg + (size_t)(chunk + GRID) * (ROWS * IN) + tid * 28, 0, 1);
    __syncthreads();

    // ================= encoder: h = relu(x@W1+b1); z = h@W2+b2 =================
    v2f ax[RT][7];                       // hoisted x A-fragments
#pragma unroll
    for (int rt = 0; rt < RT; ++rt)
#pragma unroll
      for (int t = 0; t < 7; ++t) {
        const int k0 = 4 * t + koff;
        const int row = rbase + rt * 16 + m16;
        ax[rt][t].x = sX[row * IN + k0];
        ax[rt][t].y = sX[row * IN + k0 + 1];
      }

    v8f zacc[RT][4];
#pragma unroll
    for (int rt = 0; rt < RT; ++rt)
#pragma unroll
      for (int zn = 0; zn < 4; ++zn) zacc[rt][zn] = zero8();

#pragma unroll 1
    for (int hn = 0; hn < 8; ++hn) {     // 16-col tile of h
      v8f hacc[RT] = {zero8(), zero8()};
#pragma unroll
      for (int t = 0; t < 7; ++t) {      // K = 28; one B-fragment feeds RT WMMAs
        const int k0 = 4 * t + koff;
        v2f b; const int n = hn * 16 + m16;
        b.x = sW1[k0 * HID + n]; b.y = sW1[(k0 + 1) * HID + n];
        hacc[0] = WMMA_F32(ax[0][t], b, hacc[0]);
        hacc[1] = WMMA_F32(ax[1][t], b, hacc[1]);
      }
      // bounce each tile through LDS to re-layout C/D -> A fragments
      v2f af[RT][4];
#pragma unroll
      for (int rt = 0; rt < RT; ++rt) {
        const int n = hn * 16 + m16;
        const float bias = sB1[n];
#pragma unroll
        for (int r = 0; r < 8; ++r) {
          float v = hacc[rt][r] + bias;
          v = v > 0.f ? v : 0.f;
          sS[wid * 256 + (r + 8 * half) * 16 + m16] = v;
        }
#pragma unroll
        for (int tt = 0; tt < 4; ++tt) {
          const int kl = 4 * tt + koff;
          af[rt][tt].x = sS[wid * 256 + m16 * 16 + kl];
          af[rt][tt].y = sS[wid * 256 + m16 * 16 + kl + 1];
        }
      }
#pragma unroll
      for (int tt = 0; tt < 4; ++tt) {   // feed this 16-wide k-chunk into z GEMM
        const int kg = hn * 16 + 4 * tt + koff;
#pragma unroll
        for (int zn = 0; zn < 4; ++zn) {
          v2f b; const int n = zn * 16 + m16;
          b.x = sW2[kg * LAT + n]; b.y = sW2[(kg + 1) * LAT + n];
          zacc[0][zn] = WMMA_F32(af[0][tt], b, zacc[0][zn]);
          zacc[1][zn] = WMMA_F32(af[1][tt], b, zacc[1][zn]);
        }
      }
    }

    // stage z per row-tile through LDS, hoist A-fragments + ||z||^2
    v2f afrag[RT][16];                   // z fragments, later replaced by quantized
    float z2v[RT][8];
#pragma unroll
    for (int rt = 0; rt < RT; ++rt) {
#pragma unroll
      for (int zn = 0; zn < 4; ++zn) {
        const int n = zn * 16 + m16;
        const float bz = sB2[n];
#pragma unroll
        for (int r = 0; r < 8; ++r)
          sZQ[wid * 1024 + (r + 8 * half) * LAT + n] = zacc[rt][zn][r] + bz;
      }
#pragma unroll
      for (int t = 0; t < 16; ++t) {
        const int k0 = 4 * t + koff;
        afrag[rt][t].x = sZQ[wid * 1024 + m16 * LAT + k0];
        afrag[rt][t].y = sZQ[wid * 1024 + m16 * LAT + k0 + 1];
      }
      float part = 0.f;
#pragma unroll
      for (int t = 0; t < 16; ++t)
        part += afrag[rt][t].x * afrag[rt][t].x + afrag[rt][t].y * afrag[rt][t].y;
      part += __shfl_xor(part, 16, 32);  // combine the two k-halves of row m16
      if (half == 0) sZ2[wid * 32 + rt * 16 + m16] = part;
#pragma unroll
      for (int r = 0; r < 8; ++r) z2v[rt][r] = sZ2[wid * 32 + rt * 16 + r + 8 * half];
    }

    // ============ nearest code: dist = ||z||^2 + ||e||^2 - 2 z.e ============
    float bestVal[RT][8]; int bestIdx[RT][8];
#pragma unroll
    for (int rt = 0; rt < RT; ++rt)
#pragma unroll
      for (int r = 0; r < 8; ++r) { bestVal[rt][r] = 3.4e38f; bestIdx[rt][r] = 0; }

#pragma unroll 1
    for (int ct = 0; ct < 32; ++ct) {    // 32 tiles of 16 codes
      v8f d0 = zero8(), d1 = zero8();
#pragma unroll
      for (int t = 0; t < 16; ++t) {     // K = 64; one B-fragment, two WMMAs
        const int k0 = 4 * t + koff;
        v2f b; const int c = ct * 16 + m16;
        b.x = sCBt[k0 * NC + c]; b.y = sCBt[(k0 + 1) * NC + c];
        d0 = WMMA_F32(afrag[0][t], b, d0);
        d1 = WMMA_F32(afrag[1][t], b, d1);
      }
      const int c = ct * 16 + m16;
      const float e2 = sE2[c];
#pragma unroll
      for (int r = 0; r < 8; ++r) {
        const float v0 = z2v[0][r] + e2 - 2.f * d0[r];
        if (v0 < bestVal[0][r]) { bestVal[0][r] = v0; bestIdx[0][r] = c; }
        const float v1 = z2v[1][r] + e2 - 2.f * d1[r];
        if (v1 < bestVal[1][r]) { bestVal[1][r] = v1; bestIdx[1][r] = c; }
      }
    }
    // min-reduce across the 16 lanes holding each row (ties -> lower index)
#pragma unroll
    for (int off = 8; off >= 1; off >>= 1) {
#pragma unroll
      for (int rt = 0; rt < RT; ++rt)
#pragma unroll
        for (int r = 0; r < 8; ++r) {
          const float ov = __shfl_xor(bestVal[rt][r], off, 32);
          const int   oi = __shfl_xor(bestIdx[rt][r], off, 32);
          if (ov < bestVal[rt][r] || (ov == bestVal[rt][r] && oi < bestIdx[rt][r])) {
            bestVal[rt][r] = ov; bestIdx[rt][r] = oi;
          }
        }
    }
    if (m16 == 0) {
#pragma unroll
      for (int rt = 0; rt < RT; ++rt)
#pragma unroll
        for (int r = 0; r < 8; ++r)
          sIdx[wid * 32 + rt * 16 + r + 8 * half] = bestIdx[rt][r];
    }

    // gather quantized rows straight into A-fragment registers + VQ-loss partial
    // (lane covers row m16, k in {4t+2*half, +1}: each element exactly once)
#pragma unroll
    for (int rt = 0; rt < RT; ++rt) {
      const int c = sIdx[wid * 32 + rt * 16 + m16];
#pragma unroll
      for (int t = 0; t < 16; ++t) {
        const int k0 = 4 * t + koff;
        const float q0 = sCBt[k0 * NC + c];
        const float q1 = sCBt[(k0 + 1) * NC + c];
        const float e0 = afrag[rt][t].x - q0;
        const float e1 = afrag[rt][t].y - q1;
        vqAcc += e0 * e0 + e1 * e1;
        afrag[rt][t].x = q0;             // straight-through: z_q == quantized fwd
        afrag[rt][t].y = q1;
      }
    }

    // ========== decoder: hd = relu(q@DW1+db1); recon = hd@DW2+db2 ==========
    v8f racc[RT][2];
#pragma unroll
    for (int rt = 0; rt < RT; ++rt) { racc[rt][0] = zero8(); racc[rt][1] = zero8(); }

#pragma unroll 1
    for (int hn = 0; hn < 8; ++hn) {
      v8f hacc[RT] = {zero8(), zero8()};
#pragma unroll
      for (int t = 0; t < 16; ++t) {     // K = 64; one B-fragment, two WMMAs
        const int k0 = 4 * t + koff;
        v2f b; const int n = hn * 16 + m16;
        b.x = sDW1[k0 * HID + n]; b.y = sDW1[(k0 + 1) * HID + n];
        hacc[0] = WMMA_F32(afrag[0][t], b, hacc[0]);
        hacc[1] = WMMA_F32(afrag[1][t], b, hacc[1]);
      }
      v2f af[RT][4];
#pragma unroll
      for (int rt = 0; rt < RT; ++rt) {
        const int n = hn * 16 + m16;
        const float bias = sDB1[n];
#pragma unroll
        for (int r = 0; r < 8; ++r) {
          float v = hacc[rt][r] + bias;
          v = v > 0.f ? v : 0.f;
          sS[wid * 256 + (r + 8 * half) * 16 + m16] = v;
        }
#pragma unroll
        for (int tt = 0; tt < 4; ++tt) {
          const int kl = 4 * tt + koff;
          af[rt][tt].x = sS[wid * 256 + m16 * 16 + kl];
          af[rt][tt].y = sS[wid * 256 + m16 * 16 + kl + 1];
        }
      }
#pragma unroll
      for (int tt = 0; tt < 4; ++tt) {
        const int kg = hn * 16 + 4 * tt + koff;
#pragma unroll
        for (int rn = 0; rn < 2; ++rn) {
          v2f b; const int n = rn * 16 + m16;
          b.x = sDW2[kg * 32 + n]; b.y = sDW2[(kg + 1) * 32 + n];
          racc[0][rn] = WMMA_F32(af[0][tt], b, racc[0][rn]);
          racc[1][rn] = WMMA_F32(af[1][tt], b, racc[1][rn]);
        }
      }
    }

    // recon: bounce each tile through LDS for coalesced b128 writeback + loss
#pragma unroll
    for (int rt = 0; rt < RT; ++rt) {
#pragma unroll
      for (int rn = 0; rn < 2; ++rn) {
        const int n = rn * 16 + m16;
        if (n < IN) {
          const float bb = sDB2[n];
#pragma unroll
          for (int r = 0; r < 8; ++r) {
            const int row = r + 8 * half;
            sZQ[wid * 1024 + row * IN + n] = racc[rt][rn][r] + bb;
          }
        }
      }
      const int lrow = rbase + rt * 16;                              // local row base
      const size_t gBase = ((size_t)chunk * ROWS + lrow) * IN;       // float offset
      float4* __restrict__ outv = (float4*)(recon + gBase);          // 16B aligned
#pragma unroll 1
      for (int j = lane; j < (16 * IN) / 4; j += 32) {               // 112 float4 per tile
        const float4 rv = *(const float4*)&sZQ[wid * 1024 + j * 4];
        const float4 xv = *(const float4*)&sX[lrow * IN + j * 4];
        const float d0 = rv.x - xv.x, d1 = rv.y - xv.y;
        const float d2 = rv.z - xv.z, d3 = rv.w - xv.w;
        rlAcc += d0 * d0 + d1 * d1 + d2 * d2 + d3 * d3;
        outv[j] = rv;
      }
    }
  }

  // ---- block loss reduction -> global atomics ----
#pragma unroll
  for (int off = 16; off >= 1; off >>= 1) {
    rlAcc += __shfl_xor(rlAcc, off, 32);
    vqAcc += __shfl_xor(vqAcc, off, 32);
  }
  if (lane == 0) { sRed[wid] = rlAcc; sRed[8 + wid] = vqAcc; }
  __syncthreads();
  if (tid == 0) {
    float r = 0.f, v = 0.f;
    for (int w = 0; w < 8; ++w) { r += sRed[w]; v += sRed[8 + w]; }
    atomicAdd(&acc[0], r);
    atomicAdd(&acc[1], v);
  }
}

__global__ void vqvae_zero(float* acc) {
  if (threadIdx.x < 2) acc[threadIdx.x] = 0.f;
}

__global__ void vqvae_final(const float* __restrict__ acc, float* __restrict__ out) {
  if (threadIdx.x == 0) {
    out[(size_t)BATCH * IN]     = acc[0] / (float)((size_t)BATCH * IN);
    // codebook_loss + 0.25*commitment_loss == 1.25 * mean((z-q)^2) forward
    out[(size_t)BATCH * IN + 1] = 1.25f * (acc[1] / (float)((size_t)BATCH * LAT));
  }
}

extern "C" void kernel_launch(void* const* d_in, const int* in_sizes, int n_in,
                              void* d_out, int out_size, void* d_ws, size_t ws_size,
                              hipStream_t stream) {
  (void)in_sizes; (void)n_in; (void)out_size; (void)ws_size;
  const float* x   = (const float*)d_in[0];
  const float* w1  = (const float*)d_in[1];
  const float* b1  = (const float*)d_in[2];
  const float* w2  = (const float*)d_in[3];
  const float* b2  = (const float*)d_in[4];
  const float* cb  = (const float*)d_in[5];
  const float* dw1 = (const float*)d_in[6];
  const float* db1 = (const float*)d_in[7];
  const float* dw2 = (const float*)d_in[8];
  const float* db2 = (const float*)d_in[9];
  float* out = (float*)d_out;
  float* acc = (float*)d_ws;

  vqvae_zero<<<1, 32, 0, stream>>>(acc);
  vqvae_fwd<<<GRID, 256, 0, stream>>>(x, w1, b1, w2, b2, cb, dw1, db1, dw2, db2, out, acc);
  vqvae_final<<<1, 32, 0, stream>>>(acc, out);
}